// VACLVA_953482740102
// MI455X (gfx1250) — compile-verified
//
#include <hip/hip_runtime.h>
#include <math.h>

// ---------------------------------------------------------------------------
// Problem constants (from reference)
// ---------------------------------------------------------------------------
constexpr int B_  = 16;
constexpr int DV  = 512;
constexpr int DA  = 256;
constexpr int S_  = 1024;
constexpr int K_  = 256;
constexpr int DJ  = DV + DA;          // 768
constexpr int N_  = B_ * S_;          // 16384 samples for correlation
constexpr float MU_  = 0.5f;
constexpr float EPS_ = 1e-5f;

typedef __attribute__((ext_vector_type(2))) float v2f;
typedef __attribute__((ext_vector_type(8))) float v8f;

// CDNA5 fp32 matrix core: D(16x16) = A(16x4) * B(4x16) + C
// A frag: lane<16 -> {A[m=lane][k0],A[m][k0+1]} with k0=0; lane>=16 -> k0=2
// B frag: lane<16 -> {B[k0][n=lane],B[k0+1][n]}   k0 = 2*(lane>>4)
// C/D   : vgpr r, lane -> D[m=r+8*(lane>>4)][n=lane&15]
__device__ __forceinline__ v8f wmma4(v2f a, v2f b, v8f c) {
    return __builtin_amdgcn_wmma_f32_16x16x4_f32(
        /*neg_a=*/false, a, /*neg_b=*/false, b,
        /*c_mod=*/(short)0, c, /*reuse_a=*/false, /*reuse_b=*/false);
}

#define ZERO8 {0.f,0.f,0.f,0.f,0.f,0.f,0.f,0.f}

// pair-interleaved LDS layout: element (row, n) -> (row/2)*32 + n*2 + (row&1)
// => a (row, row+1) pair for fixed n is one contiguous 8B word (ds_load_b64),
//    and lanes 0..15 / 16..31 hit disjoint bank sets.
__device__ __forceinline__ int pidx(int row, int n) {
    return (row >> 1) * 32 + n * 2 + (row & 1);
}

// ---------------------------------------------------------------------------
// Kernel 0: transpose W_hv (DV,K)->(K,DV), W_ha (DA,K)->(K,DA)
// ---------------------------------------------------------------------------
__global__ __launch_bounds__(256)
void k_transpose(const float* __restrict__ Whv, const float* __restrict__ Wha,
                 float* __restrict__ WhvT, float* __restrict__ WhaT) {
    int idx = blockIdx.x * 256 + threadIdx.x;
    if (idx < DV * K_) {
        int d = idx / K_, k = idx % K_;
        WhvT[(size_t)k * DV + d] = Whv[idx];
    }
    if (idx < DA * K_) {
        int d = idx / K_, k = idx % K_;
        WhaT[(size_t)k * DA + d] = Wha[idx];
    }
}

// ---------------------------------------------------------------------------
// Kernel 1: diag_Mv[b,s] = tanh(scale * sum_j (Wjv^T Xv)[j,s] * J[j,s])
//           diag_Ma[b,s] = tanh(scale * sum_j (Wja^T Xa)[j,s] * J[j,s])
// One block = (b, 16-column s-tile). J tile (768x16) staged once in LDS.
// ---------------------------------------------------------------------------
__global__ __launch_bounds__(256)
void k_diag(const float* __restrict__ Xv, const float* __restrict__ Xa,
            const float* __restrict__ Wjv, const float* __restrict__ Wja,
            float* __restrict__ dMv, float* __restrict__ dMa) {
    __shared__ float J[DJ * 16];          // 48 KB, pair-interleaved
    __shared__ float partV[16], partA[16];

    const int s0  = blockIdx.x * 16;
    const int b   = blockIdx.y;
    const int tid = threadIdx.x;

    for (int idx = tid; idx < DJ * 16; idx += 256) {
        int row = idx >> 4, n = idx & 15;
        float v = (row < DV)
            ? Xv[((size_t)(b * DV + row)) * S_ + s0 + n]
            : Xa[((size_t)(b * DA + (row - DV))) * S_ + s0 + n];
        J[pidx(row, n)] = v;
    }
    if (tid < 16) { partV[tid] = 0.f; partA[tid] = 0.f; }
    __syncthreads();

    const int wave = tid >> 5;
    const int lane = tid & 31;
    const int half = lane >> 4;
    const int l16  = lane & 15;

    float sumV = 0.f, sumA = 0.f;

    for (int jt = wave; jt < DJ / 16; jt += 8) {
        const int jb = jt * 16;

        // ---- v-part: P = Wjv^T (16j x DV) * Xv-tile (DV x 16s)
        v8f acc = ZERO8;
        #pragma unroll 4
        for (int kb = 0; kb < DV; kb += 4) {
            v2f a;
            a.x = Wjv[(size_t)(kb + 2 * half    ) * DJ + jb + l16];
            a.y = Wjv[(size_t)(kb + 2 * half + 1) * DJ + jb + l16];
            v2f bf = *(const v2f*)&J[(kb / 2 + half) * 32 + l16 * 2];
            acc = wmma4(a, bf, acc);
        }
        float p = 0.f;
        #pragma unroll
        for (int r = 0; r < 8; ++r) {
            int row = jb + r + 8 * half;
            p += acc[r] * J[pidx(row, l16)];
        }
        sumV += p;

        // ---- a-part: P = Wja^T (16j x DA) * Xa-tile (DA x 16s)
        v8f acc2 = ZERO8;
        #pragma unroll 4
        for (int kb = 0; kb < DA; kb += 4) {
            v2f a;
            a.x = Wja[(size_t)(kb + 2 * half    ) * DJ + jb + l16];
            a.y = Wja[(size_t)(kb + 2 * half + 1) * DJ + jb + l16];
            v2f bf = *(const v2f*)&J[(256 + kb / 2 + half) * 32 + l16 * 2]; // rows DV+kb
            acc2 = wmma4(a, bf, acc2);
        }
        p = 0.f;
        #pragma unroll
        for (int r = 0; r < 8; ++r) {
            int row = jb + r + 8 * half;
            p += acc2[r] * J[pidx(row, l16)];
        }
        sumA += p;
    }

    atomicAdd(&partV[l16], sumV);
    atomicAdd(&partA[l16], sumA);
    __syncthreads();

    if (tid < 16) {
        const float scale = 1.0f / sqrtf((float)DJ);
        dMv[b * S_ + s0 + tid] = tanhf(scale * partV[tid]);
        dMa[b * S_ + s0 + tid] = tanhf(scale * partA[tid]);
    }
}

// ---------------------------------------------------------------------------
// Kernel 2: H_v = relu(W_v Xv + W_mv * diag_Mv),  H_a = relu(W_a Xa + ...)
// Stored transposed: H[b][s][k] so each lane writes two contiguous float4.
// ---------------------------------------------------------------------------
__global__ __launch_bounds__(256)
void k_H(const float* __restrict__ Xv, const float* __restrict__ Xa,
         const float* __restrict__ Wv, const float* __restrict__ Wa,
         const float* __restrict__ Wmv, const float* __restrict__ Wma,
         const float* __restrict__ dMv, const float* __restrict__ dMa,
         float* __restrict__ Hv, float* __restrict__ Ha) {
    __shared__ float Lv[DV * 16];   // 32 KB
    __shared__ float La[DA * 16];   // 16 KB

    const int s0  = blockIdx.x * 16;
    const int b   = blockIdx.y;
    const int tid = threadIdx.x;

    for (int idx = tid; idx < DV * 16; idx += 256) {
        int row = idx >> 4, n = idx & 15;
        Lv[pidx(row, n)] = Xv[((size_t)(b * DV + row)) * S_ + s0 + n];
    }
    for (int idx = tid; idx < DA * 16; idx += 256) {
        int row = idx >> 4, n = idx & 15;
        La[pidx(row, n)] = Xa[((size_t)(b * DA + row)) * S_ + s0 + n];
    }
    __syncthreads();

    const int wave = tid >> 5;
    const int lane = tid & 31;
    const int half = lane >> 4;
    const int l16  = lane & 15;

    const float dgv = dMv[b * S_ + s0 + l16];
    const float dga = dMa[b * S_ + s0 + l16];

    for (int kt = wave; kt < K_ / 16; kt += 8) {
        // ---- H_v tile: M = k-rows, N = s-cols, K-dim = DV
        v8f acc = ZERO8;
        #pragma unroll 4
        for (int kb = 0; kb < DV; kb += 4) {
            v2f a  = *(const v2f*)(Wv + (size_t)(kt * 16 + l16) * DV + kb + 2 * half);
            v2f bf = *(const v2f*)&Lv[(kb / 2 + half) * 32 + l16 * 2];
            acc = wmma4(a, bf, acc);
        }
        float h[8];
        #pragma unroll
        for (int r = 0; r < 8; ++r) {
            int krow = kt * 16 + r + 8 * half;
            h[r] = fmaxf(acc[r] + Wmv[(size_t)krow * S_ + s0 + l16] * dgv, 0.f);
        }
        float* dst = Hv + ((size_t)(b * S_ + s0 + l16)) * K_ + kt * 16 + 8 * half;
        ((float4*)dst)[0] = make_float4(h[0], h[1], h[2], h[3]);
        ((float4*)dst)[1] = make_float4(h[4], h[5], h[6], h[7]);

        // ---- H_a tile
        v8f acc2 = ZERO8;
        #pragma unroll 4
        for (int kb = 0; kb < DA; kb += 4) {
            v2f a  = *(const v2f*)(Wa + (size_t)(kt * 16 + l16) * DA + kb + 2 * half);
            v2f bf = *(const v2f*)&La[(kb / 2 + half) * 32 + l16 * 2];
            acc2 = wmma4(a, bf, acc2);
        }
        #pragma unroll
        for (int r = 0; r < 8; ++r) {
            int krow = kt * 16 + r + 8 * half;
            h[r] = fmaxf(acc2[r] + Wma[(size_t)krow * S_ + s0 + l16] * dga, 0.f);
        }
        dst = Ha + ((size_t)(b * S_ + s0 + l16)) * K_ + kt * 16 + 8 * half;
        ((float4*)dst)[0] = make_float4(h[0], h[1], h[2], h[3]);
        ((float4*)dst)[1] = make_float4(h[4], h[5], h[6], h[7]);
    }
}

// ---------------------------------------------------------------------------
// Kernel 3: X_att = X + W_h * H, written into d_out as X_va (B, DJ, S).
// Orientation M=s, N=d so every lane stores 2 contiguous float4 of output.
// ---------------------------------------------------------------------------
__global__ __launch_bounds__(256)
void k_att(const float* __restrict__ Xv, const float* __restrict__ Xa,
           const float* __restrict__ WhvT, const float* __restrict__ WhaT,
           const float* __restrict__ Hv, const float* __restrict__ Ha,
           float* __restrict__ out) {
    __shared__ float Lv[K_ * 16];   // 16 KB  H_v tile (k-major pairs)
    __shared__ float La[K_ * 16];   // 16 KB  H_a tile

    const int s0  = blockIdx.x * 16;
    const int b   = blockIdx.y;
    const int tid = threadIdx.x;

    for (int idx = tid; idx < K_ * 16; idx += 256) {
        int n = idx >> 8, k = idx & 255;   // H stored [b][s][k]: contiguous in k
        Lv[pidx(k, n)] = Hv[((size_t)(b * S_ + s0 + n)) * K_ + k];
        La[pidx(k, n)] = Ha[((size_t)(b * S_ + s0 + n)) * K_ + k];
    }
    __syncthreads();

    const int wave = tid >> 5;
    const int lane = tid & 31;
    const int half = lane >> 4;
    const int l16  = lane & 15;

    // ---- v half: d in [0, DV)
    for (int dt = wave; dt < DV / 16; dt += 8) {
        const int db = dt * 16;
        v8f acc = ZERO8;
        #pragma unroll 4
        for (int kb = 0; kb < K_; kb += 4) {
            v2f a = *(const v2f*)&Lv[(kb / 2 + half) * 32 + l16 * 2];   // A[m=s][k]
            v2f bf;
            bf.x = WhvT[(size_t)(kb + 2 * half    ) * DV + db + l16];   // B[k][n=d]
            bf.y = WhvT[(size_t)(kb + 2 * half + 1) * DV + db + l16];
            acc = wmma4(a, bf, acc);
        }
        const float* xs = Xv + ((size_t)(b * DV + db + l16)) * S_ + s0 + 8 * half;
        float*      dst = out + ((size_t)(b * DJ + db + l16)) * S_ + s0 + 8 * half;
        float4 x0 = ((const float4*)xs)[0], x1 = ((const float4*)xs)[1];
        ((float4*)dst)[0] = make_float4(acc[0]+x0.x, acc[1]+x0.y, acc[2]+x0.z, acc[3]+x0.w);
        ((float4*)dst)[1] = make_float4(acc[4]+x1.x, acc[5]+x1.y, acc[6]+x1.z, acc[7]+x1.w);
    }

    // ---- a half: d in [0, DA), output rows DV+d
    for (int dt = wave; dt < DA / 16; dt += 8) {
        const int db = dt * 16;
        v8f acc = ZERO8;
        #pragma unroll 4
        for (int kb = 0; kb < K_; kb += 4) {
            v2f a = *(const v2f*)&La[(kb / 2 + half) * 32 + l16 * 2];
            v2f bf;
            bf.x = WhaT[(size_t)(kb + 2 * half    ) * DA + db + l16];
            bf.y = WhaT[(size_t)(kb + 2 * half + 1) * DA + db + l16];
            acc = wmma4(a, bf, acc);
        }
        const float* xs = Xa + ((size_t)(b * DA + db + l16)) * S_ + s0 + 8 * half;
        float*      dst = out + ((size_t)(b * DJ + DV + db + l16)) * S_ + s0 + 8 * half;
        float4 x0 = ((const float4*)xs)[0], x1 = ((const float4*)xs)[1];
        ((float4*)dst)[0] = make_float4(acc[0]+x0.x, acc[1]+x0.y, acc[2]+x0.z, acc[3]+x0.w);
        ((float4*)dst)[1] = make_float4(acc[4]+x1.x, acc[5]+x1.y, acc[6]+x1.z, acc[7]+x1.w);
    }
}

// ---------------------------------------------------------------------------
// Kernel 4: per-feature column sums of X_va (one block per feature d)
// ---------------------------------------------------------------------------
__global__ __launch_bounds__(256)
void k_colsum(const float* __restrict__ out, float* __restrict__ colsum) {
    const int d = blockIdx.x, tid = threadIdx.x;
    float s = 0.f;
    for (int b = 0; b < B_; ++b) {
        const float* p = out + ((size_t)(b * DJ + d)) * S_;
        for (int t = tid; t < S_; t += 256) s += p[t];
    }
    __shared__ float red[256];
    red[tid] = s; __syncthreads();
    for (int off = 128; off; off >>= 1) {
        if (tid < off) red[tid] += red[tid + off];
        __syncthreads();
    }
    if (tid == 0) colsum[d] = red[0];
}

// ---------------------------------------------------------------------------
// Kernel 5: Gram matrix G[i,j] = sum_{b,s} X_va[b,i,s] * X_va[b,j,s]
// Block = one 16x16 tile; 8 waves split the K=16384 reduction, LDS reduce.
// ---------------------------------------------------------------------------
__global__ __launch_bounds__(256)
void k_gram(const float* __restrict__ out, float* __restrict__ G) {
    constexpr int LDB = 258;               // pad: kill 16-way LDS bank conflicts
    __shared__ float LA[16 * LDB];
    __shared__ float LB[16 * LDB];

    const int ti = blockIdx.x, tj = blockIdx.y;
    const int tid  = threadIdx.x;
    const int wave = tid >> 5;
    const int lane = tid & 31;
    const int half = lane >> 4;
    const int l16  = lane & 15;

    v8f acc = ZERO8;

    for (int c = 0; c < 64; ++c) {         // 64 K-chunks of 256 (each inside one b)
        const int b  = c >> 2;
        const int s0 = (c & 3) * 256;
        __syncthreads();                   // protect previous chunk's LDS reads
        #pragma unroll
        for (int e = 0; e < 16; ++e) {
            int idx = tid + e * 256;
            int i = idx >> 8, kk = idx & 255;
            LA[i * LDB + kk] = out[((size_t)(b * DJ + ti * 16 + i)) * S_ + s0 + kk];
            LB[i * LDB + kk] = out[((size_t)(b * DJ + tj * 16 + i)) * S_ + s0 + kk];
        }
        if (c + 1 < 64) {                  // prefetch next chunk (global_prefetch_b8)
            int b2 = (c + 1) >> 2, t0 = ((c + 1) & 3) * 256;
            if (tid < 16)
                __builtin_prefetch(out + ((size_t)(b2 * DJ + ti * 16 + tid)) * S_ + t0, 0, 0);
            else if (tid < 32)
                __builtin_prefetch(out + ((size_t)(b2 * DJ + tj * 16 + (tid - 16))) * S_ + t0, 0, 0);
        }
        __syncthreads();

        const int ks0 = wave * 32;         // each wave owns 32 of the 256 K-slices
        #pragma unroll
        for (int k4 = 0; k4 < 32; k4 += 4) {
            v2f a  = *(const v2f*)&LA[l16 * LDB + ks0 + k4 + 2 * half];
            v2f bf = *(const v2f*)&LB[l16 * LDB + ks0 + k4 + 2 * half];
            acc = wmma4(a, bf, acc);
        }
    }

    __syncthreads();
    float* red = LA;                       // reuse: 8 waves x 16 x 16 = 8 KB
    #pragma unroll
    for (int r = 0; r < 8; ++r)
        red[(wave * 16 + r + 8 * half) * 16 + l16] = acc[r];
    __syncthreads();

    const int m = tid >> 4, n = tid & 15;
    float s = 0.f;
    #pragma unroll
    for (int w = 0; w < 8; ++w) s += red[(w * 16 + m) * 16 + n];
    G[(size_t)(ti * 16 + m) * DJ + tj * 16 + n] = s;
}

// ---------------------------------------------------------------------------
// Kernel 6: loss from G + colsum (single small block)
// C_ij = (G_ij - N mu_i mu_j) / (N (sig_i+eps)(sig_j+eps)), sig unbiased
// loss = sum_d (1-C_dd)^2 + MU*(sum C^2 - sum C_dd^2)
// ---------------------------------------------------------------------------
__global__ __launch_bounds__(256)
void k_loss(const float* __restrict__ G, const float* __restrict__ colsum,
            float* __restrict__ out_loss) {
    __shared__ float mu[DJ], invd[DJ];
    const int tid = threadIdx.x;
    const float Nf = (float)N_;

    for (int i = tid; i < DJ; i += 256) {
        float m = colsum[i] / Nf;
        mu[i] = m;
        float var = (G[(size_t)i * DJ + i] - Nf * m * m) / (Nf - 1.0f);
        var = fmaxf(var, 0.f);
        invd[i] = 1.0f / (sqrtf(var) + EPS_);
    }
    __syncthreads();

    float sAll = 0.f, sD1 = 0.f, sD2 = 0.f;
    for (int idx = tid; idx < DJ * DJ; idx += 256) {
        int i = idx / DJ, j = idx - i * DJ;
        float C = (G[idx] - Nf * mu[i] * mu[j]) * invd[i] * invd[j] * (1.0f / Nf);
        sAll += C * C;
        if (i == j) { float d = 1.f - C; sD1 += d * d; sD2 += C * C; }
    }

    __shared__ float rA[256], r1[256], r2[256];
    rA[tid] = sAll; r1[tid] = sD1; r2[tid] = sD2;
    __syncthreads();
    for (int off = 128; off; off >>= 1) {
        if (tid < off) { rA[tid] += rA[tid+off]; r1[tid] += r1[tid+off]; r2[tid] += r2[tid+off]; }
        __syncthreads();
    }
    if (tid == 0) out_loss[0] = r1[0] + MU_ * (rA[0] - r2[0]);
}

// ---------------------------------------------------------------------------
// Launch
// ---------------------------------------------------------------------------
extern "C" void kernel_launch(void* const* d_in, const int* in_sizes, int n_in,
                              void* d_out, int out_size, void* d_ws, size_t ws_size,
                              hipStream_t stream) {
    const float* Xv  = (const float*)d_in[0];
    const float* Xa  = (const float*)d_in[1];
    const float* Wjv = (const float*)d_in[2];
    const float* Wja = (const float*)d_in[3];
    const float* Wa  = (const float*)d_in[4];
    const float* Wv  = (const float*)d_in[5];
    const float* Wma = (const float*)d_in[6];
    const float* Wmv = (const float*)d_in[7];
    const float* Whv = (const float*)d_in[8];
    const float* Wha = (const float*)d_in[9];
    float* out = (float*)d_out;

    float* ws   = (float*)d_ws;
    float* dMv  = ws;  ws += B_ * S_;                 // 16K
    float* dMa  = ws;  ws += B_ * S_;                 // 16K
    float* Hv   = ws;  ws += (size_t)B_ * S_ * K_;    // 4M floats
    float* Ha   = ws;  ws += (size_t)B_ * S_ * K_;    // 4M floats
    float* G    = ws;  ws += DJ * DJ;                 // 589K
    float* csum = ws;  ws += DJ;
    float* WhvT = ws;  ws += K_ * DV;
    float* WhaT = ws;  ws += K_ * DA;

    k_transpose<<<dim3((DV * K_ + 255) / 256), 256, 0, stream>>>(Whv, Wha, WhvT, WhaT);
    k_diag<<<dim3(S_ / 16, B_), 256, 0, stream>>>(Xv, Xa, Wjv, Wja, dMv, dMa);
    k_H<<<dim3(S_ / 16, B_), 256, 0, stream>>>(Xv, Xa, Wv, Wa, Wmv, Wma, dMv, dMa, Hv, Ha);
    k_att<<<dim3(S_ / 16, B_), 256, 0, stream>>>(Xv, Xa, WhvT, WhaT, Hv, Ha, out);
    k_colsum<<<dim3(DJ), 256, 0, stream>>>(out, csum);
    k_gram<<<dim3(DJ / 16, DJ / 16), 256, 0, stream>>>(out, G);
    k_loss<<<1, 256, 0, stream>>>(G, csum, out + (size_t)B_ * DJ * S_);
}